// Atten_1116691497442
// MI455X (gfx1250) — compile-verified
//
#include <hip/hip_runtime.h>

typedef __attribute__((ext_vector_type(2))) float v2f;
typedef __attribute__((ext_vector_type(8))) float v8f;

#define NH 8      // heads
#define SD 32     // key size per head
#define NODEF 256 // H*C*D floats per edge/node

// order-preserving float<->uint encoding so atomicMax(uint) == float max,
// with 0 as the identity element (memset-friendly).
__device__ __forceinline__ unsigned f2ord(float f) {
    unsigned u = __float_as_uint(f);
    return (u & 0x80000000u) ? ~u : (u | 0x80000000u);
}
__device__ __forceinline__ float ord2f(unsigned u) {
    return (u & 0x80000000u) ? __uint_as_float(u & 0x7FFFFFFFu)
                             : __uint_as_float(~u);
}

// ---------------------------------------------------------------------------
// Pass 1: per-edge/head logits via V_WMMA_F32_16X16X4_F32.
// One wave handles a tile of 16 edges. A = K rows (16 edges x 32), B = Q
// columns gathered at dst (32 x 16 edges); 8 chained 16x16x4 WMMAs give a
// 16x16 f32 block whose DIAGONAL is k[e]·q[dst[e]]. Extract diag (branchless
// cndmask chain — no exec-mask cascade), scale, store logit, atomic-max into
// the per-node running max.
// ---------------------------------------------------------------------------
__global__ void pass1_logits(const float* __restrict__ k,
                             const float* __restrict__ q,
                             const int* __restrict__ dst,
                             float* __restrict__ e_ws,
                             unsigned* __restrict__ m_enc,
                             int E) {
    const int lane = threadIdx.x & 31;
    const int wave = threadIdx.x >> 5;
    const int tile = blockIdx.x * (blockDim.x >> 5) + wave;
    const int e0 = tile * 16;
    if (e0 >= E) return; // wave-uniform guard; EXEC all-1 inside

    // A-row / B-col this lane feeds (32-bit operand striping, ISA 7.12.2):
    // lanes 0-15 carry K = base+0,+1 ; lanes 16-31 carry K = base+2,+3.
    const int col   = lane & 15;
    const int khalf = (lane >> 4) << 1;
    const int ecol  = min(e0 + col, E - 1);
    const int dcol  = dst[ecol];

    // Diagonal ownership in C/D layout: i<8 -> (VGPR i, lane i);
    // i>=8 -> (VGPR i-8, lane i+16)  => valid lanes are 0-7 and 24-31.
    const bool valid = (lane < 8) || (lane >= 24);
    const int  irow  = (lane < 8) ? lane : (lane - 16);
    const int  r     = (lane < 8) ? lane : (lane - 24);
    const bool do_store = valid && (e0 + irow) < E;
    int dstore = 0;
    if (do_store) dstore = dst[e0 + irow];

    const float scale = 1.0f / 16.0f; // 1/sqrt(H*S) = 1/sqrt(256)

    for (int h = 0; h < NH; ++h) {
        const float* krow = k + ((size_t)ecol * NH + h) * SD;
        const float* qrow = q + ((size_t)dcol * NH + h) * SD;
        v8f acc = {};
#pragma unroll
        for (int kc = 0; kc < SD / 4; ++kc) {
            const int ko = kc * 4 + khalf;
            v2f a, b;
            a.x = krow[ko]; a.y = krow[ko + 1];
            b.x = qrow[ko]; b.y = qrow[ko + 1];
            acc = __builtin_amdgcn_wmma_f32_16x16x4_f32(
                false, a, false, b, (short)0, acc, false, false);
        }
        // Branchless diagonal pick: straight-line v_cndmask + adds.
        float dg = 0.0f;
#pragma unroll
        for (int j = 0; j < 8; ++j) {
            dg += (r == j) ? acc[j] : 0.0f;
        }
        if (do_store) {
            const float logit = dg * scale;
            e_ws[(size_t)(e0 + irow) * NH + h] = logit;
            atomicMax(&m_enc[(size_t)dstore * NH + h], f2ord(logit));
        }
    }
}

// ---------------------------------------------------------------------------
// Pass 2: den[n,h] = sum over edges of exp(e - m[dst]).
// ---------------------------------------------------------------------------
__global__ void pass2_den(const float* __restrict__ e_ws,
                          const unsigned* __restrict__ m_enc,
                          const int* __restrict__ dst,
                          float* __restrict__ den,
                          int EH) {
    const int t = blockIdx.x * blockDim.x + threadIdx.x;
    if (t >= EH) return;
    const int e = t >> 3;
    const int h = t & 7;
    const int d = dst[e];
    const float m  = ord2f(m_enc[(size_t)d * NH + h]); // finite: node has >=1 edge
    const float ex = __expf(e_ws[t] - m);
    unsafeAtomicAdd(&den[(size_t)d * NH + h], ex);
}

// ---------------------------------------------------------------------------
// Pass 3: one wave per edge. a = exp(e-m)/max(den,1e-9); scatter a*v[e,h,c,d]
// into out[dst]. Lane L owns 8 contiguous floats, all within head L>>2.
// ---------------------------------------------------------------------------
__global__ void pass3_scatter(const float* __restrict__ e_ws,
                              const unsigned* __restrict__ m_enc,
                              const float* __restrict__ den,
                              const int* __restrict__ dst,
                              const float* __restrict__ v,
                              float* __restrict__ out,
                              int E) {
    const int wave = (blockIdx.x * blockDim.x + threadIdx.x) >> 5;
    const int lane = threadIdx.x & 31;
    if (wave >= E) return;
    const int e = wave;
    const int h = lane >> 2;
    const int d = dst[e];
    const float m  = ord2f(m_enc[(size_t)d * NH + h]);
    const float dn = den[(size_t)d * NH + h];
    const float a  = __expf(e_ws[(size_t)e * NH + h] - m) / fmaxf(dn, 1e-9f);

    const float* vp = v   + (size_t)e * NODEF + lane * 8;
    float*       op = out + (size_t)d * NODEF + lane * 8;
    const float4 v0 = *(const float4*)(vp);
    const float4 v1 = *(const float4*)(vp + 4);
    unsafeAtomicAdd(op + 0, a * v0.x);
    unsafeAtomicAdd(op + 1, a * v0.y);
    unsafeAtomicAdd(op + 2, a * v0.z);
    unsafeAtomicAdd(op + 3, a * v0.w);
    unsafeAtomicAdd(op + 4, a * v1.x);
    unsafeAtomicAdd(op + 5, a * v1.y);
    unsafeAtomicAdd(op + 6, a * v1.z);
    unsafeAtomicAdd(op + 7, a * v1.w);
}

extern "C" void kernel_launch(void* const* d_in, const int* in_sizes, int n_in,
                              void* d_out, int out_size, void* d_ws, size_t ws_size,
                              hipStream_t stream) {
    const float* v   = (const float*)d_in[0];   // [E,H,C,D]
    const float* k   = (const float*)d_in[1];   // [E,H,S]
    const float* q   = (const float*)d_in[2];   // [N,H,S]
    const int*   dst = (const int*)d_in[3];     // [E]
    float* out = (float*)d_out;                 // [N, H*C, D] flat

    const int E = in_sizes[3];
    const int N = in_sizes[2] / (NH * SD);

    // workspace layout: e_ws[E*H] f32 | m_enc[N*H] u32 | den[N*H] f32
    float*    e_ws  = (float*)d_ws;
    unsigned* m_enc = (unsigned*)((char*)d_ws + (size_t)E * NH * sizeof(float));
    float*    den   = (float*)((char*)m_enc + (size_t)N * NH * sizeof(unsigned));

    hipMemsetAsync(m_enc, 0, (size_t)N * NH * sizeof(unsigned), stream); // identity of f2ord-max
    hipMemsetAsync(den,   0, (size_t)N * NH * sizeof(float), stream);
    hipMemsetAsync(out,   0, (size_t)out_size * sizeof(float), stream);

    const int tiles = (E + 15) / 16;            // 16 edges per wave
    pass1_logits<<<(tiles + 7) / 8, 256, 0, stream>>>(k, q, dst, e_ws, m_enc, E);

    const int EH = E * NH;
    pass2_den<<<(EH + 255) / 256, 256, 0, stream>>>(e_ws, m_enc, dst, den, EH);

    pass3_scatter<<<(E + 7) / 8, 256, 0, stream>>>(e_ws, m_enc, den, dst, v, out, E);
}